// MultiHeadAttention_78984448573623
// MI455X (gfx1250) — compile-verified
//
#include <hip/hip_runtime.h>

typedef __attribute__((ext_vector_type(2))) float v2f;
typedef __attribute__((ext_vector_type(8))) float v8f;

#define B_    4
#define S_    2048
#define D_    1024
#define H_    16
#define HD_   64
#define N3_   3072      // 3*H*HD
#define MTOT_ 8192      // B*S

static __device__ __forceinline__ v8f wmma4(v2f a, v2f b, v8f c) {
  // D = A(16x4 f32) * B(4x16 f32) + C(16x16 f32)
  return __builtin_amdgcn_wmma_f32_16x16x4_f32(false, a, false, b, (short)0, c,
                                               false, false);
}

// ---------------------------------------------------------------------------
// Kernel 1: QKV projection  C[M=8192, N=3072] = X[8192,1024] @ W[1024,3072]+b
// Block: 256 thr = 8 waves (2 M-waves x 4 N-waves). Wave tile: 32M x 64N.
// Epilogue routes columns into Q/K/V workspace laid out [B][H][S][HD].
// ---------------------------------------------------------------------------
__global__ __launch_bounds__(256) void qkv_gemm_k(
    const float* __restrict__ X, const float* __restrict__ W,
    const float* __restrict__ bias, float* __restrict__ qkv) {
  const int tid  = threadIdx.x;
  const int lane = tid & 31, wid = tid >> 5;
  const int l15  = lane & 15, hi = lane >> 4;
  const int wm   = wid & 1,  wn  = wid >> 1;
  const int m_base = blockIdx.y * 64 + wm * 32;
  const int n_base = blockIdx.x * 256 + wn * 64;

  v8f acc[2][4] = {};
  const float* Xr0 = X + (size_t)(m_base + l15) * D_;
  const float* Xr1 = X + (size_t)(m_base + 16 + l15) * D_;

  for (int k = 0; k < D_; k += 4) {
    const int kk = k + 2 * hi;                 // A/B k-pair for this lane half
    v2f a0 = *(const v2f*)(Xr0 + kk);
    v2f a1 = *(const v2f*)(Xr1 + kk);
    const float* Wr0 = W + (size_t)kk * N3_ + n_base + l15;
    const float* Wr1 = Wr0 + N3_;
    v2f bf[4];
#pragma unroll
    for (int nt = 0; nt < 4; ++nt) {
      bf[nt].x = Wr0[nt * 16];
      bf[nt].y = Wr1[nt * 16];
    }
#pragma unroll
    for (int nt = 0; nt < 4; ++nt) {
      acc[0][nt] = wmma4(a0, bf[nt], acc[0][nt]);
      acc[1][nt] = wmma4(a1, bf[nt], acc[1][nt]);
    }
  }

#pragma unroll
  for (int mt = 0; mt < 2; ++mt) {
#pragma unroll
    for (int nt = 0; nt < 4; ++nt) {
      const int n  = n_base + nt * 16 + l15;
      const float bv = bias[n];
      const int which = n >> 10, rem = n & 1023;   // 0:Q 1:K 2:V
      const int h = rem >> 6, hd = rem & 63;
#pragma unroll
      for (int j = 0; j < 8; ++j) {
        const int m = m_base + mt * 16 + 8 * hi + j;   // C layout row
        const int b = m >> 11, s = m & 2047;
        const size_t off = (size_t)which * ((size_t)B_ * H_ * S_ * HD_) +
                           (((size_t)(b * H_ + h)) * S_ + s) * HD_ + hd;
        qkv[off] = acc[mt][nt][j] + bv;
      }
    }
  }
}

// ---------------------------------------------------------------------------
// Kernel 2: flash attention. One wave per (b,h,16-query tile).
// S^T = K @ Q^T  (softmax rows land in-lane), then O^T += V^T @ P.
// ---------------------------------------------------------------------------
__global__ __launch_bounds__(256) void attn_k(
    const float* __restrict__ Q, const float* __restrict__ K,
    const float* __restrict__ V, const unsigned char* __restrict__ mask,
    float* __restrict__ out) {
  __shared__ float lds[8][16 * 68];                 // per-wave padded 16x64
  const int tid  = threadIdx.x;
  const int lane = tid & 31, wid = tid >> 5;
  const int l15  = lane & 15, hi = lane >> 4;
  const int task = blockIdx.x * 8 + wid;            // < B*H*(S/16) = 8192
  const int qt = task & 127;
  const int bh = task >> 7;
  const int h  = bh & 15, b = bh >> 4;

  const float* Qp = Q + ((size_t)bh * S_ + qt * 16) * HD_;
  const float* Kp = K + (size_t)bh * S_ * HD_;
  const float* Vp = V + (size_t)bh * S_ * HD_;
  const unsigned char* mp0 = mask + ((size_t)b * S_ + qt * 16 + l15) * S_;

  // Q^T as B-fragments (K-dim = 64 -> 16 steps), pre-scaled by 1/sqrt(HD)
  v2f qb[16];
#pragma unroll
  for (int t = 0; t < 16; ++t) {
    v2f q = *(const v2f*)(Qp + l15 * HD_ + 4 * t + 2 * hi);
    qb[t] = q * 0.125f;
  }

  v8f o[4] = {};                                    // O^T: 4 hd-tiles x 16 q
  float mrow = -3.0e38f, lrow = 0.0f;               // per-q stats (q = l15)

  for (int k0 = 0; k0 < S_; k0 += 16) {
    // ---- scores^T: rows = key (j + 8*hi), cols = q (l15) ----
    v8f s = {};
#pragma unroll
    for (int t = 0; t < 16; ++t) {
      v2f ka = *(const v2f*)(Kp + (size_t)(k0 + l15) * HD_ + 4 * t + 2 * hi);
      s = wmma4(ka, qb[t], s);
    }
    // ---- mask ----
    const unsigned char* mp = mp0 + k0 + 8 * hi;
#pragma unroll
    for (int j = 0; j < 8; ++j)
      if (!mp[j]) s[j] = -1.0e9f;
    // ---- online softmax (row = this lane's q) ----
    float tmax = s[0];
#pragma unroll
    for (int j = 1; j < 8; ++j) tmax = fmaxf(tmax, s[j]);
    tmax = fmaxf(tmax, __shfl_xor(tmax, 16, 32));
    const float mnew = fmaxf(mrow, tmax);
    const float c = __expf(mrow - mnew);
    float p[8], po[8], rs = 0.0f;
#pragma unroll
    for (int j = 0; j < 8; ++j) { p[j] = __expf(s[j] - mnew); rs += p[j]; }
    rs += __shfl_xor(rs, 16, 32);
    lrow = lrow * c + rs;
    mrow = mnew;
#pragma unroll
    for (int mt = 0; mt < 4; ++mt) o[mt] *= c;      // c is lane-uniform in q
    // ---- exchange halves: lane gets all 16 keys for its q ----
#pragma unroll
    for (int j = 0; j < 8; ++j) po[j] = __shfl_xor(p[j], 16, 32);
    // B-fragments of P (register selects only, constant indices after unroll)
    v2f pb[4];
#pragma unroll
    for (int t = 0; t < 4; ++t) {
#pragma unroll
      for (int j = 0; j < 2; ++j) {
        const int kA = 4 * t + j;        // key if hi==0
        const int kB = 4 * t + 2 + j;    // key if hi==1
        const float vA = (kA < 8) ? p[kA] : po[kA - 8];
        const float vB = (kB < 8) ? po[kB] : p[kB - 8];
        const float val = hi ? vB : vA;
        if (j == 0) pb[t].x = val; else pb[t].y = val;
      }
    }
    // ---- O^T += V^T @ P  (coalesced V loads) ----
#pragma unroll
    for (int t = 0; t < 4; ++t) {
      const float* Vr0 = Vp + (size_t)(k0 + 4 * t + 2 * hi) * HD_ + l15;
      const float* Vr1 = Vr0 + HD_;
#pragma unroll
      for (int mt = 0; mt < 4; ++mt) {
        v2f va;
        va.x = Vr0[mt * 16];
        va.y = Vr1[mt * 16];
        o[mt] = wmma4(va, pb[t], o[mt]);
      }
    }
  }

  // ---- normalize, transpose via LDS, coalesced store to [B,S,D] ----
  const float inv = 1.0f / lrow;
  float* L = &lds[wid][0];
#pragma unroll
  for (int mt = 0; mt < 4; ++mt) {
    o[mt] *= inv;
#pragma unroll
    for (int j = 0; j < 8; ++j) {
      const int hd = mt * 16 + 8 * hi + j;          // O^T row
      L[l15 * 68 + hd] = o[mt][j];
    }
  }
  __syncthreads();
  float* outp = out + ((size_t)b * S_ + qt * 16) * D_ + h * HD_;
  for (int idx = lane; idx < 16 * 64; idx += 32) {
    const int q = idx >> 6, hd = idx & 63;
    outp[(size_t)q * D_ + hd] = L[q * 68 + hd];
  }
}

// ---------------------------------------------------------------------------
extern "C" void kernel_launch(void* const* d_in, const int* in_sizes, int n_in,
                              void* d_out, int out_size, void* d_ws,
                              size_t ws_size, hipStream_t stream) {
  (void)in_sizes; (void)n_in; (void)out_size; (void)ws_size;
  const float* X            = (const float*)d_in[0];
  const unsigned char* mask = (const unsigned char*)d_in[1];
  const float* W            = (const float*)d_in[2];
  const float* bias         = (const float*)d_in[3];
  float* out = (float*)d_out;
  float* qkv = (float*)d_ws;               // 3 * B*H*S*HD floats (~96 MB)

  dim3 g1(N3_ / 256, MTOT_ / 64);          // (12, 128)
  qkv_gemm_k<<<g1, 256, 0, stream>>>(X, W, bias, qkv);

  const size_t seg = (size_t)B_ * H_ * S_ * HD_;
  attn_k<<<dim3(B_ * H_ * (S_ / 16) / 8), 256, 0, stream>>>(
      qkv, qkv + seg, qkv + 2 * seg, mask, out);
}